// N_Gram_Embedding_87522843558257
// MI455X (gfx1250) — compile-verified
//
#include <hip/hip_runtime.h>

// N-gram embedding mean: out[w, :] = (1/cnt[w]) * sum_k emb[ids[w,k], :], id==0 -> zero row.
// One wave32 per word. CDNA5 path: branchless burst of 24 async global->LDS gathers
// (ASYNCcnt, in-order completion) with staged s_wait_asynccnt pipelining the reduction.
// Padded slots gather the WGP$-hot row 0 and are zeroed by a scalar fmac mask.

#define B_SZ   16
#define S_SZ   2048
#define WORDS  (B_SZ * S_SZ)     // 32768
#define K_NG   24
#define E_DIM  128
#define WPB    4                 // waves per block
#define THREADS (WPB * 32)

__global__ __launch_bounds__(THREADS) void ngram_embed_kernel(
    const int*   __restrict__ word_idx,      // [B*S]
    const int*   __restrict__ ngram_ids,     // [V, K]
    const int*   __restrict__ ngram_counts,  // [V]
    const float* __restrict__ emb_table,     // [NG, E]
    float*       __restrict__ out)           // [B*S, E]
{
    // 4 waves * 24 rows * 128 f32 = 48 KB LDS staging
    __shared__ __align__(16) float buf[WPB * K_NG * E_DIM];

    const int lane   = (int)(threadIdx.x & 31u);
    const int waveId = __builtin_amdgcn_readfirstlane((int)(threadIdx.x >> 5));
    const int wlin   = __builtin_amdgcn_readfirstlane((int)blockIdx.x * WPB + waveId);
    if (wlin >= WORDS) return;

    // Uniform scalar loads (SMEM path)
    const int wv  = word_idx[wlin];
    const int cnt = ngram_counts[wv];

    // One coalesced vector load: lanes 0..23 fetch this word's 24 ngram ids.
    const int slot = (lane < K_NG) ? lane : (K_NG - 1);   // clamp, no OOB on last word
    int idv = ngram_ids[wv * K_NG + slot];
    if (lane >= K_NG) idv = 0;

    // Broadcast each id to an SGPR exactly once; stays live across the waits.
    int ids[K_NG];
    #pragma unroll
    for (int k = 0; k < K_NG; ++k)
        ids[k] = __builtin_amdgcn_readlane(idv, k);

    const uint32_t lane16  = (uint32_t)lane * 16u;                      // 16 B per lane
    float* wbuf = &buf[waveId * (K_NG * E_DIM)];
    const uint32_t ldsBase = (uint32_t)(uintptr_t)wbuf + lane16;        // LDS byte addr
    const uint64_t embPtr  = (uint64_t)(uintptr_t)emb_table;

    // ---- Branchless burst: 24 async b128 gathers, back-to-back ----
    // id==0 rows fetch table row 0 (one hot 512B line, served from WGP$) and are
    // masked out at accumulate time. ASYNCcnt-tracked; no data VGPRs in flight.
    #pragma unroll
    for (int k = 0; k < K_NG; ++k) {
        const uint32_t voff = (uint32_t)ids[k] * (E_DIM * 4u) + lane16; // < 2^27
        const uint32_t ldsa = ldsBase + (uint32_t)k * (E_DIM * 4u);
        asm volatile("global_load_async_to_lds_b128 %0, %1, %2"
                     :: "v"(ldsa), "v"(voff), "s"(embPtr)
                     : "memory");
    }

    // ---- Reduce from LDS with staged waits (async loads complete in order):
    //      consume rows 0-7 while rows 8-23 are still landing, etc. ----
    float4 acc = make_float4(0.f, 0.f, 0.f, 0.f);
    #pragma unroll
    for (int k = 0; k < K_NG; ++k) {
        if (k == 0)  asm volatile("s_wait_asynccnt 16" ::: "memory");
        if (k == 8)  asm volatile("s_wait_asynccnt 8"  ::: "memory");
        if (k == 16) asm volatile("s_wait_asynccnt 0"  ::: "memory");
        const float m = (ids[k] != 0) ? 1.0f : 0.0f;     // scalar (SGPR) mask
        const float4 v = *(const float4*)(wbuf + k * E_DIM + lane * 4);
        acc.x = fmaf(m, v.x, acc.x);
        acc.y = fmaf(m, v.y, acc.y);
        acc.z = fmaf(m, v.z, acc.z);
        acc.w = fmaf(m, v.w, acc.w);
    }

    const float c = (float)cnt;   // cnt >= 1 always
    float4 r;
    r.x = acc.x / c; r.y = acc.y / c; r.z = acc.z / c; r.w = acc.w / c;
    *(float4*)(out + (size_t)wlin * E_DIM + lane * 4) = r;
}

extern "C" void kernel_launch(void* const* d_in, const int* in_sizes, int n_in,
                              void* d_out, int out_size, void* d_ws, size_t ws_size,
                              hipStream_t stream) {
    (void)in_sizes; (void)n_in; (void)d_ws; (void)ws_size; (void)out_size;
    const int*   word_idx     = (const int*)d_in[0];
    const int*   ngram_ids    = (const int*)d_in[1];
    const int*   ngram_counts = (const int*)d_in[2];
    const float* emb_table    = (const float*)d_in[3];
    float*       out          = (float*)d_out;

    dim3 grid(WORDS / WPB);   // 8192 blocks, 4 waves each -> one wave per word
    ngram_embed_kernel<<<grid, THREADS, 0, stream>>>(
        word_idx, ngram_ids, ngram_counts, emb_table, out);
}